// JAWSNetwork_3908420239529
// MI455X (gfx1250) — compile-verified
//
#include <hip/hip_runtime.h>
#include <hip/hip_bf16.h>

#define NNODES 100000
#define NEDGES 1600000
#define F      128
#define GROWS  80   // rows per GEMM block: 100000 = 1250 * 80

typedef __attribute__((ext_vector_type(2))) float v2f;
typedef __attribute__((ext_vector_type(8))) float v8f;

// ---------------------------------------------------------------------------
// Async global->LDS staging (CDNA5 GLOBAL_LOAD_ASYNC_TO_LDS_B128, ASYNCcnt).
// Builtin prototype (from hipcc diagnostic): param0 = global (AS1) int4*
// (gcc vector_size(16)), param1 = LDS (AS3) int4*, then imm offset, imm cpol.
// ---------------------------------------------------------------------------
typedef int v4i_vs __attribute__((__vector_size__(16)));
typedef v4i_vs __attribute__((address_space(1)))* gptr16_t;
typedef v4i_vs __attribute__((address_space(3)))* lptr16_t;

__device__ __forceinline__ void async_copy16(float* lds, const float* g) {
#if __has_builtin(__builtin_amdgcn_global_load_async_to_lds_b128)
  __builtin_amdgcn_global_load_async_to_lds_b128(
      (gptr16_t)(uintptr_t)g, (lptr16_t)(uintptr_t)lds, 0, 0);
#else
  *(float4*)lds = *(const float4*)g;
#endif
}

__device__ __forceinline__ void wait_async() {
#if __has_builtin(__builtin_amdgcn_global_load_async_to_lds_b128)
#if __has_builtin(__builtin_amdgcn_s_wait_asynccnt)
  __builtin_amdgcn_s_wait_asynccnt(0);
#else
  asm volatile("s_wait_asynccnt 0x0" ::: "memory");
#endif
#endif
}

// ---------------------------------------------------------------------------
// Degree / normalization
// ---------------------------------------------------------------------------
__global__ void k_deg_init(float* __restrict__ deg) {
  int i = blockIdx.x * blockDim.x + threadIdx.x;
  if (i < NNODES) deg[i] = 1.0f;  // self-loop
}

__global__ void k_deg_count(const int* __restrict__ ei, float* __restrict__ deg) {
  int e = blockIdx.x * blockDim.x + threadIdx.x;
  if (e < NEDGES) atomicAdd(&deg[ei[NEDGES + e]], 1.0f);  // col = target
}

__global__ void k_dis(float* __restrict__ deg) {
  int i = blockIdx.x * blockDim.x + threadIdx.x;
  if (i < NNODES) deg[i] = rsqrtf(deg[i]);  // deg >= 1 always
}

// ---------------------------------------------------------------------------
// GEMM: out[N,128] = X[N,128] @ W[128,128] via V_WMMA_F32_16X16X4_F32
// Block = 80 rows (5 row-tiles) x 128 cols. 8 waves, one 16-col tile each;
// each wave loops the 5 row-tiles, reusing its B fragment 5x per k-step.
// X block + full W staged in LDS (async) once per block.
// ---------------------------------------------------------------------------
__global__ __launch_bounds__(256) void k_gemm128(const float* __restrict__ X,
                                                 const float* __restrict__ W,
                                                 float* __restrict__ out) {
  __shared__ float sX[GROWS * 132];   // 132-dword padded rows (bank spread)
  __shared__ float sW[128 * 132];

  const int t = threadIdx.x;
  const size_t blockRow = (size_t)blockIdx.x * GROWS;

  // Stage X block: 80x128 floats = 2560 float4, 10 per thread (coalesced).
#pragma unroll
  for (int i = 0; i < 10; ++i) {
    int idx = t + 256 * i;              // float4 index
    int r = idx >> 5;
    int c = (idx & 31) << 2;
    async_copy16(&sX[r * 132 + c], X + (blockRow + r) * F + c);
  }
  // Stage full W: 128x128 floats = 4096 float4, 16 per thread (coalesced).
#pragma unroll
  for (int i = 0; i < 16; ++i) {
    int idx = t + 256 * i;
    int r = idx >> 5;
    int c = (idx & 31) << 2;
    async_copy16(&sW[r * 132 + c], W + (size_t)r * F + c);
  }
  wait_async();
  __syncthreads();

  const int lane = t & 31;
  const int wave = t >> 5;
  const int m    = lane & 15;     // A row / B col within tile
  const int half = lane >> 4;     // 0: K{0,1}; 1: K{2,3}
  const int kb   = half * 2;
  const int col  = wave * 16 + m; // output column owned by this lane

  v8f acc[5] = {};
  for (int k = 0; k < 128; k += 4) {
    v2f b;
    b.x = sW[(k + kb) * 132 + col];
    b.y = sW[(k + kb + 1) * 132 + col];
#pragma unroll
    for (int rt = 0; rt < 5; ++rt) {
      v2f a;
      a.x = sX[(rt * 16 + m) * 132 + k + kb];
      a.y = sX[(rt * 16 + m) * 132 + k + kb + 1];
      acc[rt] = __builtin_amdgcn_wmma_f32_16x16x4_f32(false, a, false, b,
                                                      (short)0, acc[rt],
                                                      false, false);
    }
  }

  // D 16x16 f32: VGPR r holds M = r + 8*half, N = lane&15
#pragma unroll
  for (int rt = 0; rt < 5; ++rt) {
#pragma unroll
    for (int r = 0; r < 8; ++r) {
      out[(blockRow + rt * 16 + r + 8 * half) * F + col] = acc[rt][r];
    }
  }
}

// ---------------------------------------------------------------------------
// Aggregation: out = self-loop term, then scatter-add edges, then bias+relu
// ---------------------------------------------------------------------------
__global__ void k_self_init(const float* __restrict__ t128,
                            const float* __restrict__ dis,
                            float* __restrict__ out) {
  int tid = blockIdx.x * blockDim.x + threadIdx.x;   // N*32 threads (float4 each)
  int node = tid >> 5;
  float d = dis[node];
  float s = d * d;
  float4 v = ((const float4*)t128)[tid];
  ((float4*)out)[tid] = make_float4(v.x * s, v.y * s, v.z * s, v.w * s);
}

__global__ __launch_bounds__(256) void k_scatter128(const int* __restrict__ ei,
                                                    const float* __restrict__ dis,
                                                    const float* __restrict__ t128,
                                                    float* __restrict__ out) {
  int gid  = blockIdx.x * blockDim.x + threadIdx.x;
  int e    = gid >> 5;          // one wave32 per edge
  int lane = gid & 31;          // float4 per lane -> 128 features
  int r = ei[e];                // source
  int c = ei[NEDGES + e];       // target
  float norm = dis[r] * dis[c];
  float4 v = ((const float4*)(t128 + (size_t)r * F))[lane];
  float* dst = out + (size_t)c * F + lane * 4;
  atomicAdd(dst + 0, v.x * norm);   // resolves in 192MB L2
  atomicAdd(dst + 1, v.y * norm);
  atomicAdd(dst + 2, v.z * norm);
  atomicAdd(dst + 3, v.w * norm);
}

__global__ void k_bias_relu(float* __restrict__ out, const float* __restrict__ b) {
  int tid = blockIdx.x * blockDim.x + threadIdx.x;   // N*128 threads
  float v = out[tid] + b[tid & (F - 1)];
  out[tid] = v > 0.0f ? v : 0.0f;
}

// ---------------------------------------------------------------------------
// Projection head: p[N,2] = [h1|h2|h3] @ Wp(384x2), one wave32 per node
// ---------------------------------------------------------------------------
__global__ __launch_bounds__(256) void k_proj(const float* __restrict__ h1,
                                              const float* __restrict__ h2,
                                              const float* __restrict__ h3,
                                              const float* __restrict__ Wp,
                                              float* __restrict__ p) {
  int gid  = blockIdx.x * blockDim.x + threadIdx.x;
  int node = gid >> 5;
  int lane = gid & 31;
  const float* rows[3] = {h1 + (size_t)node * F, h2 + (size_t)node * F,
                          h3 + (size_t)node * F};
  float a0 = 0.f, a1 = 0.f;
#pragma unroll
  for (int s = 0; s < 3; ++s) {
#pragma unroll
    for (int f0 = 0; f0 < F; f0 += 32) {
      int f = f0 + lane;
      float v = rows[s][f];               // coalesced across lanes
      a0 += v * Wp[(s * F + f) * 2 + 0];
      a1 += v * Wp[(s * F + f) * 2 + 1];
    }
  }
#pragma unroll
  for (int off = 16; off > 0; off >>= 1) {
    a0 += __shfl_xor(a0, off, 32);
    a1 += __shfl_xor(a1, off, 32);
  }
  if (lane == 0) {
    p[node * 2 + 0] = a0;
    p[node * 2 + 1] = a1;
  }
}

__global__ void k_out_self_init(const float* __restrict__ p,
                                const float* __restrict__ dis,
                                float* __restrict__ y) {
  int i = blockIdx.x * blockDim.x + threadIdx.x;
  if (i >= NNODES) return;
  float s = dis[i] * dis[i];
  y[i * 2 + 0] = p[i * 2 + 0] * s;
  y[i * 2 + 1] = p[i * 2 + 1] * s;
}

__global__ void k_out_scatter(const int* __restrict__ ei, const float* __restrict__ dis,
                              const float* __restrict__ p, float* __restrict__ y) {
  int e = blockIdx.x * blockDim.x + threadIdx.x;
  if (e >= NEDGES) return;
  int r = ei[e];
  int c = ei[NEDGES + e];
  float norm = dis[r] * dis[c];
  atomicAdd(&y[c * 2 + 0], p[r * 2 + 0] * norm);
  atomicAdd(&y[c * 2 + 1], p[r * 2 + 1] * norm);
}

__global__ void k_softmax(const float* __restrict__ y, const float* __restrict__ bp,
                          float* __restrict__ out) {
  int i = blockIdx.x * blockDim.x + threadIdx.x;
  if (i >= NNODES) return;
  float a = y[i * 2 + 0] + bp[0];
  float b = y[i * 2 + 1] + bp[1];
  float m = a > b ? a : b;
  float e0 = __expf(a - m);
  float e1 = __expf(b - m);
  float inv = 1.0f / (e0 + e1);
  out[i * 2 + 0] = e0 * inv;
  out[i * 2 + 1] = e1 * inv;
}

// ---------------------------------------------------------------------------
extern "C" void kernel_launch(void* const* d_in, const int* in_sizes, int n_in,
                              void* d_out, int out_size, void* d_ws, size_t ws_size,
                              hipStream_t stream) {
  const float* x  = (const float*)d_in[0];
  const int*   ei = (const int*)d_in[1];
  const float* W0 = (const float*)d_in[2];
  const float* b0 = (const float*)d_in[3];
  const float* W1 = (const float*)d_in[4];
  const float* b1 = (const float*)d_in[5];
  const float* W2 = (const float*)d_in[6];
  const float* b2 = (const float*)d_in[7];
  const float* Wp = (const float*)d_in[8];
  const float* bp = (const float*)d_in[9];
  float* out = (float*)d_out;

  // Workspace carve-up (floats). Everything fully overwritten each call.
  float* ws  = (float*)d_ws;
  size_t o   = 0;
  float* dis = ws + o; o += NNODES;                    // deg -> dis (in place)
  float* t   = ws + o; o += (size_t)NNODES * F;        // pre-aggregation features
  float* h1  = ws + o; o += (size_t)NNODES * F;
  float* h2  = ws + o; o += (size_t)NNODES * F;
  float* h3  = ws + o; o += (size_t)NNODES * F;
  float* p   = ws + o; o += (size_t)NNODES * 2;
  float* y   = ws + o; o += (size_t)NNODES * 2;

  const int B = 256;
  const int gN    = (NNODES + B - 1) / B;      // 391
  const int gE    = NEDGES / B;                // 6250
  const int gGemm = NNODES / GROWS;            // 1250
  const int gSelf = NNODES * 32 / B;           // 12500
  const int gScat = NEDGES * 32 / B;           // 200000
  const int gBias = NNODES * F / B;            // 50000
  const int gProj = NNODES * 32 / B;           // 12500

  // --- normalization ---
  k_deg_init<<<gN, B, 0, stream>>>(dis);
  k_deg_count<<<gE, B, 0, stream>>>(ei, dis);
  k_dis<<<gN, B, 0, stream>>>(dis);

  // --- layer 1 ---
  k_gemm128<<<gGemm, B, 0, stream>>>(x, W0, t);
  k_self_init<<<gSelf, B, 0, stream>>>(t, dis, h1);
  k_scatter128<<<gScat, B, 0, stream>>>(ei, dis, t, h1);
  k_bias_relu<<<gBias, B, 0, stream>>>(h1, b0);

  // --- layer 2 ---
  k_gemm128<<<gGemm, B, 0, stream>>>(h1, W1, t);
  k_self_init<<<gSelf, B, 0, stream>>>(t, dis, h2);
  k_scatter128<<<gScat, B, 0, stream>>>(ei, dis, t, h2);
  k_bias_relu<<<gBias, B, 0, stream>>>(h2, b1);

  // --- layer 3 ---
  k_gemm128<<<gGemm, B, 0, stream>>>(h2, W2, t);
  k_self_init<<<gSelf, B, 0, stream>>>(t, dis, h3);
  k_scatter128<<<gScat, B, 0, stream>>>(ei, dis, t, h3);
  k_bias_relu<<<gBias, B, 0, stream>>>(h3, b2);

  // --- projection head ---
  k_proj<<<gProj, B, 0, stream>>>(h1, h2, h3, Wp, p);
  k_out_self_init<<<gN, B, 0, stream>>>(p, dis, y);
  k_out_scatter<<<gE, B, 0, stream>>>(ei, dis, p, y);
  k_softmax<<<gN, B, 0, stream>>>(y, bp, out);
}